// HyperGNN_68942815036073
// MI455X (gfx1250) — compile-verified
//
#include <hip/hip_runtime.h>
#include <hip/hip_bf16.h>

typedef __attribute__((ext_vector_type(2))) float v2f;
typedef __attribute__((ext_vector_type(8))) float v8f;

#define D_DIM 128

// ---------------------------------------------------------------------------
// Kernel 1: zero-init the output accumulator (atomics require clean zeros).
// ---------------------------------------------------------------------------
__global__ __launch_bounds__(256) void hgnn_zero(float4* __restrict__ out, int n4) {
    int i = blockIdx.x * blockDim.x + threadIdx.x;
    if (i < n4) out[i] = make_float4(0.f, 0.f, 0.f, 0.f);
}

// ---------------------------------------------------------------------------
// Kernel 2: h = X @ W^T via V_WMMA_F32_16X16X4_F32 (fp32-precision WMMA).
// Block = 256 threads = 8 wave32; wave w owns column tile [16w, 16w+16),
// block b owns row tile [16b, 16b+16). 32 chained WMMAs over K=128.
//
// Per-lane layouts (ISA 7.12.2, 32-bit operands, wave32):
//   A 16x4 : a[v] = A[lane%16][k + (lane>=16 ? 2:0) + v]
//   B 4x16 : b[v] = B[k + (lane>=16 ? 2:0) + v][lane%16],  B[kk][n] = W[n][kk]
//   D 16x16: acc[r] -> row (r + (lane>=16 ? 8:0)), col lane%16
// ---------------------------------------------------------------------------
__global__ __launch_bounds__(256) void hgnn_gemm_wmma(
    const float* __restrict__ X, const float* __restrict__ W,
    float* __restrict__ h, int N) {
    const int wave  = threadIdx.x >> 5;
    const int lane  = threadIdx.x & 31;
    const int row0  = blockIdx.x << 4;
    const int col0  = wave << 4;
    const int m     = lane & 15;
    const int khalf = (lane >> 4) << 1;       // 0 for lanes 0-15, 2 for 16-31

    int arow = row0 + m;
    if (arow >= N) arow = N - 1;              // clamp loads for tail tile
    const float* __restrict__ Xr = X + (size_t)arow * D_DIM;
    const float* __restrict__ Wr = W + (size_t)(col0 + m) * D_DIM;  // row n of W == col n of B

    v8f acc = {};
#pragma unroll
    for (int k = 0; k < D_DIM; k += 4) {
        const float2 af = *(const float2*)(Xr + k + khalf);
        const float2 bf = *(const float2*)(Wr + k + khalf);
        v2f a; a.x = af.x; a.y = af.y;
        v2f b; b.x = bf.x; b.y = bf.y;
        acc = __builtin_amdgcn_wmma_f32_16x16x4_f32(
            /*neg_a=*/false, a, /*neg_b=*/false, b,
            /*c_mod=*/(short)0, acc, /*reuse_a=*/false, /*reuse_b=*/false);
    }

    const int mbase = row0 + ((lane < 16) ? 0 : 8);
#pragma unroll
    for (int r = 0; r < 8; ++r) {
        const int mr = mbase + r;
        if (mr < N) h[(size_t)mr * D_DIM + col0 + m] = acc[r];
    }
}

// ---------------------------------------------------------------------------
// Kernel 3: fused gated SpMM scatter. One wave32 per edge:
//   out[rows[g][e]] += sigmoid(alpha[g]) * vals[g][e] * h[cols[g][e]]
// 32 lanes x float4 = the full 128-float row (b128 gather, 4x f32 atomics).
// h and out are L2-resident (25.6 MB each vs 192 MB L2).
// ---------------------------------------------------------------------------
__global__ __launch_bounds__(256) void hgnn_spmm(
    const float* __restrict__ h,
    const int* __restrict__ rows, const int* __restrict__ cols,
    const float* __restrict__ vals, const float* __restrict__ alpha,
    float* __restrict__ out, int E, int totalWaves) {
    const int gid  = blockIdx.x * blockDim.x + threadIdx.x;
    const int wave = gid >> 5;
    const int lane = gid & 31;
    if (wave >= totalWaves) return;

    const int g = wave / E;
    const int e = wave - g * E;
    const size_t idx = (size_t)g * E + e;

    const int   r = rows[idx];
    const int   c = cols[idx];
    const float v = vals[idx];
    const float a = alpha[g];
    const float gate = 1.0f / (1.0f + __expf(-a));
    const float w = gate * v;

    const float4 x = ((const float4*)(h + (size_t)c * D_DIM))[lane];
    float* dst = out + (size_t)r * D_DIM + lane * 4;
    atomicAdd(dst + 0, w * x.x);
    atomicAdd(dst + 1, w * x.y);
    atomicAdd(dst + 2, w * x.z);
    atomicAdd(dst + 3, w * x.w);
}

// ---------------------------------------------------------------------------
extern "C" void kernel_launch(void* const* d_in, const int* in_sizes, int n_in,
                              void* d_out, int out_size, void* d_ws, size_t ws_size,
                              hipStream_t stream) {
    const float* X     = (const float*)d_in[0];   // [N, 128]
    const float* W     = (const float*)d_in[1];   // [128, 128]
    const float* alpha = (const float*)d_in[2];   // [G]
    const int*   rows  = (const int*)  d_in[3];   // [G, E]
    const int*   cols  = (const int*)  d_in[4];   // [G, E]
    const float* vals  = (const float*)d_in[5];   // [G, E]
    float*       out   = (float*)d_out;           // [N, 128]

    const int N = in_sizes[0] / D_DIM;
    const int G = in_sizes[2];
    const int E = in_sizes[3] / G;

    float* h = (float*)d_ws;                      // [N, 128] scratch (25.6 MB)

    // 1) out = 0
    const int n4 = (N * D_DIM) / 4;
    hgnn_zero<<<(n4 + 255) / 256, 256, 0, stream>>>((float4*)out, n4);

    // 2) h = X @ W^T  (WMMA fp32)
    hgnn_gemm_wmma<<<(N + 15) / 16, 256, 0, stream>>>(X, W, h, N);

    // 3) gated sparse scatter-accumulate
    const long long totalWaves = (long long)G * E;        // 2.4M
    const long long blocks = (totalWaves * 32 + 255) / 256;
    hgnn_spmm<<<(unsigned)blocks, 256, 0, stream>>>(h, rows, cols, vals, alpha,
                                                    out, E, (int)totalWaves);
}